// GraphAttentionLayer_64252710748687
// MI455X (gfx1250) — compile-verified
//
#include <hip/hip_runtime.h>
#include <hip/hip_bf16.h>
#include <math.h>

#define B_    8
#define N_    2048
#define CIN   256
#define COUT  64
#define ALPHA 0.2f
#define NEG_BIG (-9.0e15f)
#define KCH   256            // K-chunk staged in LDS for kernel 4 (64 KB)

typedef float v2f __attribute__((ext_vector_type(2)));
typedef float v8f __attribute__((ext_vector_type(8)));

// Stage a [rows x COUT] f32 panel (rows even) into LDS in K-pair-interleaved
// layout:  dst[(k/2)*COUT + c] (as float2) = { src[k][c], src[k+1][c] }.
// All global reads and LDS writes are b128, fully coalesced.
__device__ __forceinline__ void stage_interleaved(float* __restrict__ dst,
                                                  const float* __restrict__ src,
                                                  int rows) {
    const int units = (rows >> 1) * (COUT / 4);        // (k-pair, 4-col group)
    for (int u = threadIdx.x; u < units; u += 256) {
        const int kp = u >> 4;                         // COUT/4 == 16 groups
        const int cg = (u & 15) << 2;                  // first col of group
        const float4 ra = *(const float4*)(src + (size_t)(2 * kp)     * COUT + cg);
        const float4 rb = *(const float4*)(src + (size_t)(2 * kp + 1) * COUT + cg);
        float4 lo = { ra.x, rb.x, ra.y, rb.y };
        float4 hi = { ra.z, rb.z, ra.w, rb.w };
        float* d = dst + ((size_t)kp * COUT + cg) * 2;
        *(float4*)(d)     = lo;
        *(float4*)(d + 4) = hi;
    }
}

// ---------------------------------------------------------------------------
// Kernel 1: h = input @ W      rows = B*N (16384), K = 256, cols = 64
// W (64 KB) staged into LDS (interleaved) once per block; one wave computes a
// 16-row x 64-col tile with 4 f32 WMMA accumulators.
// ---------------------------------------------------------------------------
__global__ void gat_gemm_h(const float* __restrict__ inp,
                           const float* __restrict__ W,
                           float* __restrict__ h) {
    extern __shared__ float Wl[];            // CIN*COUT floats = 64 KB

    stage_interleaved(Wl, W, CIN);
    __syncthreads();

    const int wave = threadIdx.x >> 5;
    const int lane = threadIdx.x & 31;
    const int half = lane >> 4;              // 0: lanes 0-15, 1: lanes 16-31
    const int lr   = lane & 15;
    const int rowtile = blockIdx.x * 8 + wave;   // 0..1023
    const int r0 = rowtile * 16;

    v8f acc0 = {}, acc1 = {}, acc2 = {}, acc3 = {};
    const float* arow = inp + (size_t)(r0 + lr) * CIN + 2 * half;

    for (int k = 0; k < CIN; k += 4) {
        // A frag: lanes 0-15 hold A[lr][k..k+1], lanes 16-31 hold A[lr][k+2..k+3]
        v2f a = *(const v2f*)(arow + k);
        // B frag: single ds_load_b64 per tile (interleaved layout)
        const float* wp = &Wl[((k >> 1) + half) * (COUT * 2) + lr * 2];
        v2f b0 = *(const v2f*)(wp +  0);     // cols  0-15
        v2f b1 = *(const v2f*)(wp + 32);     // cols 16-31
        v2f b2 = *(const v2f*)(wp + 64);     // cols 32-47
        v2f b3 = *(const v2f*)(wp + 96);     // cols 48-63
        acc0 = __builtin_amdgcn_wmma_f32_16x16x4_f32(false, a, false, b0, (short)0, acc0, false, false);
        acc1 = __builtin_amdgcn_wmma_f32_16x16x4_f32(false, a, false, b1, (short)0, acc1, false, false);
        acc2 = __builtin_amdgcn_wmma_f32_16x16x4_f32(false, a, false, b2, (short)0, acc2, false, false);
        acc3 = __builtin_amdgcn_wmma_f32_16x16x4_f32(false, a, false, b3, (short)0, acc3, false, false);
    }

    // C/D layout: VGPR v, lanes 0-15 -> M=v, lanes 16-31 -> M=v+8, N=lr
    float* hout = h + (size_t)r0 * COUT;
#pragma unroll
    for (int v = 0; v < 8; ++v) {
        const int row = v + 8 * half;
        hout[(size_t)row * COUT + lr +  0] = acc0[v];
        hout[(size_t)row * COUT + lr + 16] = acc1[v];
        hout[(size_t)row * COUT + lr + 32] = acc2[v];
        hout[(size_t)row * COUT + lr + 48] = acc3[v];
    }
}

// ---------------------------------------------------------------------------
// Kernel 2: f = h . a1 , g = h . a2  (per row)
// ---------------------------------------------------------------------------
__global__ void gat_fg(const float* __restrict__ h,
                       const float* __restrict__ a,
                       float* __restrict__ f,
                       float* __restrict__ g) {
    const int row = blockIdx.x * blockDim.x + threadIdx.x;
    if (row >= B_ * N_) return;
    const float* hr = h + (size_t)row * COUT;
    float fv = 0.f, gv = 0.f;
#pragma unroll 8
    for (int c = 0; c < COUT; ++c) {
        const float hv = hr[c];
        fv += hv * a[c];
        gv += hv * a[COUT + c];
    }
    f[row] = fv;
    g[row] = gv;
}

// ---------------------------------------------------------------------------
// Kernel 3: masked leaky-relu scores + row softmax -> attention (into d_out)
// One 256-thread block per (b,i) row; 8 columns per thread.  Streaming-bound:
// reads adj row (8 KB int32), writes attention row (8 KB f32).
// ---------------------------------------------------------------------------
__global__ void gat_softmax(const int* __restrict__ adj,
                            const float* __restrict__ f,
                            const float* __restrict__ g,
                            float* __restrict__ att) {
    const int bi  = blockIdx.x;      // b*N + i
    const int b   = bi >> 11;        // / N_
    const int tid = threadIdx.x;

    __shared__ float redm[8];
    __shared__ float reds[8];

    const float  fi     = f[bi];
    const int*   adjrow = adj + (size_t)bi * N_;
    const float* gb     = g + (size_t)b * N_;

    float sc[8];
    float m = NEG_BIG;
#pragma unroll
    for (int q = 0; q < 8; ++q) {
        const int j = tid + q * 256;
        float e = fi + gb[j];
        e = (e >= 0.f) ? e : ALPHA * e;
        const float s = (adjrow[j] < 1) ? e : NEG_BIG;
        sc[q] = s;
        m = fmaxf(m, s);
    }
    // wave32 max reduction
#pragma unroll
    for (int off = 16; off >= 1; off >>= 1)
        m = fmaxf(m, __shfl_xor(m, off, 32));
    if ((tid & 31) == 0) redm[tid >> 5] = m;
    __syncthreads();
    float mall = redm[0];
#pragma unroll
    for (int w = 1; w < 8; ++w) mall = fmaxf(mall, redm[w]);

    float ex[8];
    float s = 0.f;
#pragma unroll
    for (int q = 0; q < 8; ++q) {
        ex[q] = __expf(sc[q] - mall);   // masked -> exp(-huge) == 0
        s += ex[q];
    }
#pragma unroll
    for (int off = 16; off >= 1; off >>= 1)
        s += __shfl_xor(s, off, 32);
    if ((tid & 31) == 0) reds[tid >> 5] = s;
    __syncthreads();
    float stot = 0.f;
#pragma unroll
    for (int w = 0; w < 8; ++w) stot += reds[w];
    const float inv = 1.f / stot;

    float* arow = att + (size_t)bi * N_;
#pragma unroll
    for (int q = 0; q < 8; ++q)
        arow[tid + q * 256] = ex[q] * inv;
}

// ---------------------------------------------------------------------------
// Kernel 4: h' = attention @ h, then ELU.  One wave per 16-row tile; all 8
// waves of a block share one batch, so the B matrix (h_b) is staged through
// LDS in 64 KB K-chunks, K-pair interleaved so every B fragment is a single
// aligned ds_load_b64.  Attention rows stream from global with prefetch.
// ---------------------------------------------------------------------------
__global__ void gat_gemm_out(const float* __restrict__ att,
                             const float* __restrict__ h,
                             float* __restrict__ out) {
    extern __shared__ float Hl[];            // KCH*COUT floats = 64 KB

    const int wave = threadIdx.x >> 5;
    const int lane = threadIdx.x & 31;
    const int half = lane >> 4;
    const int lr   = lane & 15;
    const int rowtile = blockIdx.x * 8 + wave;   // 0..1023
    const int b  = rowtile >> 7;                 // 128 tiles per batch, 16 blocks/batch
    const int r0 = rowtile * 16;                 // global row in [0, B*N)

    const float* arow = att + (size_t)(r0 + lr) * N_ + 2 * half;
    const float* hb   = h + (size_t)b * N_ * COUT;

    v8f acc0 = {}, acc1 = {}, acc2 = {}, acc3 = {};

    for (int nc = 0; nc < N_ / KCH; ++nc) {      // 8 chunks of K
        stage_interleaved(Hl, hb + (size_t)nc * KCH * COUT, KCH);
        __syncthreads();

        // prefetch next chunk of this lane's attention row
        if (nc + 1 < N_ / KCH)
            __builtin_prefetch(arow + (nc + 1) * KCH, 0, 1);

        for (int kk = 0; kk < KCH; kk += 4) {
            v2f a = *(const v2f*)(arow + nc * KCH + kk);
            const float* hp = &Hl[((kk >> 1) + half) * (COUT * 2) + lr * 2];
            v2f b0 = *(const v2f*)(hp +  0);
            v2f b1 = *(const v2f*)(hp + 32);
            v2f b2 = *(const v2f*)(hp + 64);
            v2f b3 = *(const v2f*)(hp + 96);
            acc0 = __builtin_amdgcn_wmma_f32_16x16x4_f32(false, a, false, b0, (short)0, acc0, false, false);
            acc1 = __builtin_amdgcn_wmma_f32_16x16x4_f32(false, a, false, b1, (short)0, acc1, false, false);
            acc2 = __builtin_amdgcn_wmma_f32_16x16x4_f32(false, a, false, b2, (short)0, acc2, false, false);
            acc3 = __builtin_amdgcn_wmma_f32_16x16x4_f32(false, a, false, b3, (short)0, acc3, false, false);
        }
        __syncthreads();
    }

    float* orow = out + (size_t)r0 * COUT;
#pragma unroll
    for (int v = 0; v < 8; ++v) {
        const int row = v + 8 * half;
        float x0 = acc0[v], x1 = acc1[v], x2 = acc2[v], x3 = acc3[v];
        x0 = (x0 > 0.f) ? x0 : (__expf(x0) - 1.f);
        x1 = (x1 > 0.f) ? x1 : (__expf(x1) - 1.f);
        x2 = (x2 > 0.f) ? x2 : (__expf(x2) - 1.f);
        x3 = (x3 > 0.f) ? x3 : (__expf(x3) - 1.f);
        orow[(size_t)row * COUT + lr +  0] = x0;
        orow[(size_t)row * COUT + lr + 16] = x1;
        orow[(size_t)row * COUT + lr + 32] = x2;
        orow[(size_t)row * COUT + lr + 48] = x3;
    }
}

// ---------------------------------------------------------------------------
extern "C" void kernel_launch(void* const* d_in, const int* in_sizes, int n_in,
                              void* d_out, int out_size, void* d_ws, size_t ws_size,
                              hipStream_t stream) {
    const float* inp = (const float*)d_in[0];   // (B, N, CIN)
    const int*   adj = (const int*)d_in[1];     // (B, N, N)
    const float* W   = (const float*)d_in[2];   // (CIN, COUT)
    const float* a   = (const float*)d_in[3];   // (2*COUT, 1)

    float* out_elu = (float*)d_out;                                // B*N*COUT
    float* out_att = (float*)d_out + (size_t)B_ * N_ * COUT;       // B*N*N

    float* h = (float*)d_ws;                                       // B*N*COUT
    float* f = h + (size_t)B_ * N_ * COUT;                         // B*N
    float* g = f + (size_t)B_ * N_;                                // B*N

    const size_t lds_bytes = (size_t)KCH * COUT * sizeof(float);   // 64 KB

    // 1) h = X @ W : 1024 row tiles, 8 waves/block, W staged in LDS
    gat_gemm_h<<<128, 256, lds_bytes, stream>>>(inp, W, h);
    // 2) f, g per row
    gat_fg<<<(B_ * N_ + 255) / 256, 256, 0, stream>>>(h, a, f, g);
    // 3) masked softmax -> attention (second output region)
    gat_softmax<<<B_ * N_, 256, 0, stream>>>(adj, f, g, out_att);
    // 4) h' = attention @ h, ELU (first output region), h_b chunks via LDS
    gat_gemm_out<<<128, 256, lds_bytes, stream>>>(out_att, h, out_elu);
}